// OIAFuser_18433999635105
// MI455X (gfx1250) — compile-verified
//
#include <hip/hip_runtime.h>
#include <hip/hip_bf16.h>

// ---------------------------------------------------------------------------
// OIAFuser forward for gfx1250 (MI455X).
// All dense layers (Y = act(X @ W^T + b)) run through V_WMMA_F32_16X16X4_F32:
// fp32 end-to-end (matches the fp32 reference numerics) while exercising the
// CDNA5 matrix pipe. Workload is ~5 GFLOP / fits in L2 -> latency bound, so
// fp32 WMMA throughput is more than sufficient.
//
// GEMM shapes here guarantee M % 16 == 0, N % 64 == 0, K % 4 == 0 (the lone
// K=10 layer is zero-padded to K=12), so the WMMA GEMM has no bounds checks:
// straight-line b64 vector loads feeding 4 WMMAs per K-step.
// ---------------------------------------------------------------------------

typedef __attribute__((ext_vector_type(2))) float v2f;
typedef __attribute__((ext_vector_type(8))) float v8f;

#define BTOK 256     // B*T
#define NHP  1024
#define NOP  512
#define DMD  512
#define INTERD 256
#define TAU_ 0.05f

enum { ACT_NONE = 0, ACT_RELU = 1, ACT_GELU = 2, ACT_SIGMOID = 3 };

// ---------------------------------------------------------------------------
// GEMM: Y(M,N) = act(X(M,K) @ W(N,K)^T + bias(N))
// Requirements: M%16==0, N%64==0, K%4==0, K even, 8B-aligned buffers.
// One wave computes a 16x64 strip: 1 A fragment x 4 B fragments -> 4 WMMAs
// per 4-wide K step. fp32 WMMA VGPR layout (ISA 7.12.2):
//   A 16x4 : VGPR0 = K+0 (lanes 0-15) / K+2 (lanes 16-31); VGPR1 = K+1 / K+3
//   B 4x16 : same K split by lane half, N = lane & 15
//   C 16x16: VGPR i -> M = i (lanes 0-15) / i+8 (lanes 16-31), N = lane & 15
// ---------------------------------------------------------------------------
template <int ACT>
__global__ __launch_bounds__(32) void gemm_xwT_wmma(
    const float* __restrict__ X, const float* __restrict__ W,
    const float* __restrict__ bias, float* __restrict__ Y,
    int N, int K) {
  const int lane = threadIdx.x & 31;
  const int half = lane >> 4;
  const int l16  = lane & 15;
  const int tm = blockIdx.y;   // M/16 tiles
  const int tn = blockIdx.x;   // N/64 tiles

  const float* xr  = X + (size_t)(tm * 16 + l16) * K;
  const float* wr0 = W + (size_t)(tn * 64 +  0 + l16) * K;
  const float* wr1 = W + (size_t)(tn * 64 + 16 + l16) * K;
  const float* wr2 = W + (size_t)(tn * 64 + 32 + l16) * K;
  const float* wr3 = W + (size_t)(tn * 64 + 48 + l16) * K;

  __builtin_prefetch(xr, 0, 3);
  __builtin_prefetch(wr0, 0, 3);
  __builtin_prefetch(wr2, 0, 3);

  v8f acc0 = {0.f, 0.f, 0.f, 0.f, 0.f, 0.f, 0.f, 0.f};
  v8f acc1 = acc0, acc2 = acc0, acc3 = acc0;

  const int koff = half * 2;
#pragma unroll 2
  for (int k0 = 0; k0 < K; k0 += 4) {
    const int ka = k0 + koff;
    const v2f a  = *(const v2f*)(xr  + ka);
    const v2f b0 = *(const v2f*)(wr0 + ka);
    const v2f b1 = *(const v2f*)(wr1 + ka);
    const v2f b2 = *(const v2f*)(wr2 + ka);
    const v2f b3 = *(const v2f*)(wr3 + ka);
    acc0 = __builtin_amdgcn_wmma_f32_16x16x4_f32(false, a, false, b0, (short)0, acc0, false, false);
    acc1 = __builtin_amdgcn_wmma_f32_16x16x4_f32(false, a, false, b1, (short)0, acc1, false, false);
    acc2 = __builtin_amdgcn_wmma_f32_16x16x4_f32(false, a, false, b2, (short)0, acc2, false, false);
    acc3 = __builtin_amdgcn_wmma_f32_16x16x4_f32(false, a, false, b3, (short)0, acc3, false, false);
  }

  const int rowbase = tm * 16 + half * 8;
#pragma unroll
  for (int j = 0; j < 4; ++j) {
    const v8f acc = (j == 0) ? acc0 : (j == 1) ? acc1 : (j == 2) ? acc2 : acc3;
    const int col = tn * 64 + j * 16 + l16;
    const float bv = bias[col];
#pragma unroll
    for (int i = 0; i < 8; ++i) {
      float v = acc[i] + bv;
      if (ACT == ACT_RELU)    v = fmaxf(v, 0.f);
      if (ACT == ACT_GELU)    v = 0.5f * v * (1.f + erff(v * 0.70710678118654752f));
      if (ACT == ACT_SIGMOID) v = 1.f / (1.f + expf(-v));
      Y[(size_t)(rowbase + i) * N + col] = v;
    }
  }
}

// zero-pad rows: dst(rows, dk) <- src(rows, sk), cols >= sk zeroed
__global__ void pad_rows_k(const float* __restrict__ src, float* __restrict__ dst,
                           int rows, int sk, int dk) {
  const int total = rows * dk;
  int idx = blockIdx.x * blockDim.x + threadIdx.x;
  if (idx >= total) return;
  const int r = idx / dk, c = idx % dk;
  dst[idx] = (c < sk) ? src[r * sk + c] : 0.f;
}

// ---------------------------------------------------------------------------
// Conv stack (direct convolution; tiny fraction of total work)
// ---------------------------------------------------------------------------
__global__ void conv1_relu_k(const float* __restrict__ in, const float* __restrict__ w,
                             const float* __restrict__ b, float* __restrict__ out) {
  const int total = BTOK * 16 * 48 * 48;
  int idx = blockIdx.x * blockDim.x + threadIdx.x;
  if (idx >= total) return;
  const int x = idx % 48, y = (idx / 48) % 48, c = (idx / (48 * 48)) % 16,
            n = idx / (48 * 48 * 16);
  const float* im = in + (size_t)n * 96 * 96;
  const float* wc = w + c * 25;
  float s = b[c];
  for (int ky = 0; ky < 5; ++ky) {
    const int iy = y * 2 - 2 + ky;
    if (iy < 0 || iy >= 96) continue;
    for (int kx = 0; kx < 5; ++kx) {
      const int ix = x * 2 - 2 + kx;
      if (ix < 0 || ix >= 96) continue;
      s += im[iy * 96 + ix] * wc[ky * 5 + kx];
    }
  }
  out[idx] = fmaxf(s, 0.f);
}

__global__ void conv3x3_s2_relu_k(const float* __restrict__ in, const float* __restrict__ w,
                                  const float* __restrict__ b, float* __restrict__ out,
                                  int Cin, int Cout, int Hin, int Hout) {
  const int total = BTOK * Cout * Hout * Hout;
  int idx = blockIdx.x * blockDim.x + threadIdx.x;
  if (idx >= total) return;
  const int x = idx % Hout, y = (idx / Hout) % Hout,
            co = (idx / (Hout * Hout)) % Cout, n = idx / (Hout * Hout * Cout);
  float s = b[co];
  for (int ci = 0; ci < Cin; ++ci) {
    const float* im = in + ((size_t)n * Cin + ci) * Hin * Hin;
    const float* wk = w + ((size_t)co * Cin + ci) * 9;
    for (int ky = 0; ky < 3; ++ky) {
      const int iy = 2 * y - 1 + ky;
      if (iy < 0 || iy >= Hin) continue;
      for (int kx = 0; kx < 3; ++kx) {
        const int ix = 2 * x - 1 + kx;
        if (ix < 0 || ix >= Hin) continue;
        s += im[iy * Hin + ix] * wk[ky * 3 + kx];
      }
    }
  }
  out[idx] = fmaxf(s, 0.f);
}

__global__ void avgpool36_k(const float* __restrict__ in, float* __restrict__ out) {
  const int total = BTOK * 128;
  int idx = blockIdx.x * blockDim.x + threadIdx.x;
  if (idx >= total) return;
  const float* p = in + (size_t)idx * 36;
  float s = 0.f;
  for (int i = 0; i < 36; ++i) s += p[i];
  out[idx] = s * (1.f / 36.f);
}

// ---------------------------------------------------------------------------
// Project points, bilinear sample overlap map (grid_sample align_corners=True,
// zeros padding, then clip [0,1])
// ---------------------------------------------------------------------------
__global__ void proj_sample_k(const float* __restrict__ xyz, const float* __restrict__ ov,
                              const float* __restrict__ Km, const float* __restrict__ bbox,
                              float* __restrict__ sout, int P) {
  const int total = BTOK * P;
  int idx = blockIdx.x * blockDim.x + threadIdx.x;
  if (idx >= total) return;
  const int bt = idx / P;
  const float* pt = xyz + (size_t)idx * 3;
  const float fx = Km[0], cx = Km[2], fy = Km[4], cy = Km[5];
  const float Z = fmaxf(pt[2], 1e-6f);
  float u = fx * (pt[0] / Z) + cx;
  float v = fy * (pt[1] / Z) + cy;
  const float* bb = bbox + bt * 4;
  const float side = fmaxf(bb[2], bb[3]);
  const float ntlx = bb[0] + bb[2] * 0.5f - side * 0.5f;
  const float ntly = bb[1] + bb[3] * 0.5f - side * 0.5f;
  const float sc = 96.f / side;
  u = (u - ntlx) * sc;
  v = (v - ntly) * sc;
  const float x0 = floorf(u), y0 = floorf(v);
  const float wx1 = u - x0, wx0 = 1.f - wx1, wy1 = v - y0, wy0 = 1.f - wy1;
  const float* im = ov + (size_t)bt * 96 * 96;
  auto g = [&](float xi, float yi) -> float {
    const float valid = (xi >= 0.f && xi <= 95.f && yi >= 0.f && yi <= 95.f) ? 1.f : 0.f;
    const int xc = (int)fminf(fmaxf(xi, 0.f), 95.f);
    const int yc = (int)fminf(fmaxf(yi, 0.f), 95.f);
    return im[yc * 96 + xc] * valid;
  };
  const float val = g(x0, y0) * wx0 * wy0 + g(x0 + 1.f, y0) * wx1 * wy0 +
                    g(x0, y0 + 1.f) * wx0 * wy1 + g(x0 + 1.f, y0 + 1.f) * wx1 * wy1;
  sout[idx] = fminf(fmaxf(val, 0.f), 1.f);
}

// ---------------------------------------------------------------------------
// Interaction geometry. Only the first 10 features survive [:, :10], so the
// top-k / s_o / w_o / mean_rel / mean_dist paths of the reference are dead.
// ---------------------------------------------------------------------------
__device__ __forceinline__ float blockReduceSum256(float v, float* red) {
  const int tid = threadIdx.x;
  red[tid] = v;
  __syncthreads();
  for (int s = 128; s > 0; s >>= 1) {
    if (tid < s) red[tid] += red[tid + s];
    __syncthreads();
  }
  const float r = red[0];
  __syncthreads();
  return r;
}

__global__ __launch_bounds__(256) void rowmin_k(const float* __restrict__ hxyz,
                                                const float* __restrict__ oxyz,
                                                float* __restrict__ dmin,
                                                int* __restrict__ idxo) {
  __shared__ float sox[NOP], soy[NOP], soz[NOP], soq[NOP];
  const int bt = blockIdx.x;
  const float* ob = oxyz + (size_t)bt * NOP * 3;
  for (int i = threadIdx.x; i < NOP; i += 256) {
    const float X = ob[i * 3], Y = ob[i * 3 + 1], Z = ob[i * 3 + 2];
    sox[i] = X; soy[i] = Y; soz[i] = Z; soq[i] = X * X + Y * Y + Z * Z;
  }
  __syncthreads();
  const float* hb = hxyz + (size_t)bt * NHP * 3;
  for (int h = threadIdx.x; h < NHP; h += 256) {
    const float hx = hb[h * 3], hy = hb[h * 3 + 1], hz = hb[h * 3 + 2];
    const float hq = hx * hx + hy * hy + hz * hz;
    float best = 3.0e38f;
    int bi = 0;
    for (int o = 0; o < NOP; ++o) {
      const float d2 = hq + soq[o] - 2.f * (hx * sox[o] + hy * soy[o] + hz * soz[o]);
      if (d2 < best) { best = d2; bi = o; }
    }
    dmin[(size_t)bt * NHP + h] = sqrtf(fmaxf(best, 1e-12f));
    idxo[(size_t)bt * NHP + h] = bi;
  }
}

__global__ __launch_bounds__(256) void colmin_mean_k(const float* __restrict__ hxyz,
                                                     const float* __restrict__ oxyz,
                                                     float* __restrict__ feats) {
  __shared__ float shx[NHP], shy[NHP], shz[NHP], shq[NHP];
  __shared__ float red[256];
  const int bt = blockIdx.x;
  const float* hb = hxyz + (size_t)bt * NHP * 3;
  for (int i = threadIdx.x; i < NHP; i += 256) {
    const float X = hb[i * 3], Y = hb[i * 3 + 1], Z = hb[i * 3 + 2];
    shx[i] = X; shy[i] = Y; shz[i] = Z; shq[i] = X * X + Y * Y + Z * Z;
  }
  __syncthreads();
  const float* ob = oxyz + (size_t)bt * NOP * 3;
  float part = 0.f;
  for (int o = threadIdx.x; o < NOP; o += 256) {
    const float ox = ob[o * 3], oy = ob[o * 3 + 1], oz = ob[o * 3 + 2];
    const float oq = ox * ox + oy * oy + oz * oz;
    float best = 3.0e38f;
    for (int h = 0; h < NHP; ++h) {
      const float d2 = oq + shq[h] - 2.f * (ox * shx[h] + oy * shy[h] + oz * shz[h]);
      best = fminf(best, d2);
    }
    part += sqrtf(fmaxf(best, 1e-12f));
  }
  part = blockReduceSum256(part, red);
  if (threadIdx.x == 0) feats[bt * 10 + 9] = part / (float)NOP;
}

__global__ __launch_bounds__(256) void feats_k(const float* __restrict__ dmin,
                                               const int* __restrict__ idxo,
                                               const float* __restrict__ s_h,
                                               const float* __restrict__ hxyz,
                                               const float* __restrict__ oxyz,
                                               float* __restrict__ feats) {
  __shared__ float sv[NHP];
  __shared__ float red[256];
  const int bt = blockIdx.x, tid = threadIdx.x;
  const float* dr = dmin + (size_t)bt * NHP;
  const float* hb = hxyz + (size_t)bt * NHP * 3;
  const float* ob = oxyz + (size_t)bt * NOP * 3;
  float wsum = 0.f, dxs = 0.f, dys = 0.f, dzs = 0.f;
  for (int h = tid; h < NHP; h += 256) {
    const float d = dr[h];
    sv[h] = d;
    wsum += expf(-d / TAU_) * s_h[(size_t)bt * NHP + h];
    const int io = idxo[(size_t)bt * NHP + h];
    const float vx = ob[io * 3]     - hb[h * 3];
    const float vy = ob[io * 3 + 1] - hb[h * 3 + 1];
    const float vz = ob[io * 3 + 2] - hb[h * 3 + 2];
    const float nrm = sqrtf(fmaxf(vx * vx + vy * vy + vz * vz, 1e-6f));
    dxs += vx / nrm; dys += vy / nrm; dzs += vz / nrm;
  }
  __syncthreads();
  // bitonic sort ascending of sv[0..1023]
  for (unsigned k = 2; k <= NHP; k <<= 1) {
    for (unsigned j = k >> 1; j > 0; j >>= 1) {
      for (unsigned i = tid; i < NHP; i += 256) {
        const unsigned ixj = i ^ j;
        if (ixj > i) {
          const bool up = ((i & k) == 0);
          const float a = sv[i], b = sv[ixj];
          if (up ? (a > b) : (a < b)) { sv[i] = b; sv[ixj] = a; }
        }
      }
      __syncthreads();
    }
  }
  float s205 = 0.f, s512 = 0.f, s819 = 0.f, stot = 0.f;
  for (int i = tid; i < NHP; i += 256) {
    const float v = sv[i];
    stot += v;
    if (i < 205) s205 += v;
    if (i < 512) s512 += v;
    if (i < 819) s819 += v;
  }
  stot = blockReduceSum256(stot, red);
  s205 = blockReduceSum256(s205, red);
  s512 = blockReduceSum256(s512, red);
  s819 = blockReduceSum256(s819, red);
  wsum = blockReduceSum256(wsum, red);
  dxs  = blockReduceSum256(dxs, red);
  dys  = blockReduceSum256(dys, red);
  dzs  = blockReduceSum256(dzs, red);
  if (tid == 0) {
    float* f = feats + bt * 10;
    f[0] = stot / (float)NHP;
    f[1] = sv[0];
    f[2] = s205 / 205.f;
    f[3] = s512 / 512.f;
    f[4] = s819 / 819.f;
    f[5] = wsum / (float)NHP;
    f[6] = dxs / (float)NHP;
    f[7] = dys / (float)NHP;
    f[8] = dzs / (float)NHP;
    // f[9] written by colmin_mean_k
  }
}

// ---------------------------------------------------------------------------
// Misc elementwise / small kernels
// ---------------------------------------------------------------------------
__global__ void concat_k(const float* __restrict__ a, const float* __restrict__ b,
                         float* __restrict__ y) {
  const int total = BTOK * 256;
  int idx = blockIdx.x * blockDim.x + threadIdx.x;
  if (idx >= total) return;
  const int d = idx % 256, m = idx / 256;
  y[idx] = (d < 128) ? a[m * 128 + d] : b[m * 128 + (d - 128)];
}

// attention with sequence length 4 (batch_first=False quirk: L=B=4, N=T=64)
__global__ void attn_small_k(const float* __restrict__ qkv, float* __restrict__ o,
                             int Dm, int nhead) {
  const int dh = Dm / nhead;
  const int total = 64 * nhead * 4;
  int idx = blockIdx.x * blockDim.x + threadIdx.x;
  if (idx >= total) return;
  const int l = idx & 3;
  const int hh = (idx >> 2) % nhead;
  const int n = idx / (4 * nhead);
  const int S3 = 3 * Dm;
  const float* q = qkv + (size_t)(l * 64 + n) * S3 + hh * dh;
  float att[4];
  const float scale = rsqrtf((float)dh);
  float mx = -3.0e38f;
  for (int m = 0; m < 4; ++m) {
    const float* kp = qkv + (size_t)(m * 64 + n) * S3 + Dm + hh * dh;
    float dot = 0.f;
    for (int d = 0; d < dh; ++d) dot += q[d] * kp[d];
    att[m] = dot * scale;
    mx = fmaxf(mx, att[m]);
  }
  float se = 0.f;
  for (int m = 0; m < 4; ++m) { att[m] = expf(att[m] - mx); se += att[m]; }
  const float inv = 1.f / se;
  float* op = o + (size_t)(l * 64 + n) * Dm + hh * dh;
  for (int d = 0; d < dh; ++d) {
    float s = 0.f;
    for (int m = 0; m < 4; ++m)
      s += att[m] * qkv[(size_t)(m * 64 + n) * S3 + 2 * Dm + hh * dh + d];
    op[d] = s * inv;
  }
}

// y = LN(x + res) * g + b  (res nullable; in-place y==x allowed)
__global__ __launch_bounds__(256) void ln_res_k(const float* __restrict__ x,
                                                const float* __restrict__ res,
                                                const float* __restrict__ g,
                                                const float* __restrict__ b,
                                                float* __restrict__ y, int Dm) {
  __shared__ float red[256];
  const int row = blockIdx.x, tid = threadIdx.x;
  const float* xr = x + (size_t)row * Dm;
  const float* rr = res ? res + (size_t)row * Dm : nullptr;
  float s = 0.f;
  for (int d = tid; d < Dm; d += 256) s += xr[d] + (rr ? rr[d] : 0.f);
  s = blockReduceSum256(s, red);
  const float mean = s / (float)Dm;
  float v = 0.f;
  for (int d = tid; d < Dm; d += 256) {
    const float t = xr[d] + (rr ? rr[d] : 0.f) - mean;
    v += t * t;
  }
  v = blockReduceSum256(v, red);
  const float inv = rsqrtf(v / (float)Dm + 1e-5f);
  for (int d = tid; d < Dm; d += 256) {
    const float t = xr[d] + (rr ? rr[d] : 0.f);
    y[(size_t)row * Dm + d] = (t - mean) * inv * g[d] + b[d];
  }
}

__global__ void centers_k(const float* __restrict__ ox, float* __restrict__ ctr) {
  const int total = BTOK * 3;
  int idx = blockIdx.x * blockDim.x + threadIdx.x;
  if (idx >= total) return;
  const int d = idx % 3, bt = idx / 3;
  const float* p = ox + (size_t)bt * NOP * 3 + d;
  float s = 0.f;
  for (int i = 0; i < NOP; ++i) s += p[i * 3];
  ctr[idx] = s / (float)NOP;
}

__global__ __launch_bounds__(256) void amag_k(const float* __restrict__ ox,
                                              const float* __restrict__ ctr,
                                              float* __restrict__ amag) {
  __shared__ float red[256];
  const int bt = blockIdx.x, tid = threadIdx.x;
  const int t = bt % 64;
  float part = 0.f;
  if (t >= 2) {
    const float* p0 = ox + (size_t)bt * NOP * 3;
    const float* p1 = p0 - NOP * 3;
    const float* p2 = p1 - NOP * 3;
    const float c0x = ctr[bt * 3], c0y = ctr[bt * 3 + 1], c0z = ctr[bt * 3 + 2];
    const float c1x = ctr[(bt - 1) * 3], c1y = ctr[(bt - 1) * 3 + 1], c1z = ctr[(bt - 1) * 3 + 2];
    const float c2x = ctr[(bt - 2) * 3], c2y = ctr[(bt - 2) * 3 + 1], c2z = ctr[(bt - 2) * 3 + 2];
    for (int i = tid; i < NOP; i += 256) {
      const float ax = (p0[i * 3]     - c0x) - 2.f * (p1[i * 3]     - c1x) + (p2[i * 3]     - c2x);
      const float ay = (p0[i * 3 + 1] - c0y) - 2.f * (p1[i * 3 + 1] - c1y) + (p2[i * 3 + 1] - c2y);
      const float az = (p0[i * 3 + 2] - c0z) - 2.f * (p1[i * 3 + 2] - c1z) + (p2[i * 3 + 2] - c2z);
      part += sqrtf(ax * ax + ay * ay + az * az);
    }
  }
  part = blockReduceSum256(part, red);
  if (tid == 0) amag[bt] = part / (float)NOP;
}

__global__ __launch_bounds__(64) void gbad_k(const float* __restrict__ amag,
                                             float* __restrict__ gbad) {
  __shared__ float rmax[64], rsum[64];
  const int bcol = blockIdx.x, t = threadIdx.x;
  const float a = amag[bcol * 64 + t];
  rmax[t] = a; rsum[t] = a;
  __syncthreads();
  for (int s = 32; s > 0; s >>= 1) {
    if (t < s) {
      rmax[t] = fmaxf(rmax[t], rmax[t + s]);
      rsum[t] += rsum[t + s];
    }
    __syncthreads();
  }
  const float amax = rmax[0], amean = rsum[0] / 64.f;
  const bool shaky = (amax - amean) / (amean + 1e-6f) >= 0.4f;
  const bool high = a >= 0.4f * amax;
  gbad[bcol * 64 + t] = (shaky && high) ? 0.8f : 0.1f;
}

__global__ void film_mod_k(const float* __restrict__ hf, const float* __restrict__ of,
                           const float* __restrict__ fh, const float* __restrict__ fo,
                           const float* __restrict__ gbad, float* __restrict__ hm,
                           float* __restrict__ om) {
  const int total = BTOK * DMD;
  int idx = blockIdx.x * blockDim.x + threadIdx.x;
  if (idx >= total) return;
  const int d = idx % DMD, m = idx / DMD;
  const float gh = fh[(size_t)m * 2 * DMD + d], bh = fh[(size_t)m * 2 * DMD + DMD + d];
  const float go = fo[(size_t)m * 2 * DMD + d], bo = fo[(size_t)m * 2 * DMD + DMD + d];
  const float g = gbad[m];
  const float h = hf[idx], o = of[idx];
  hm[idx] = h + gh * h + bh;
  om[idx] = (1.f - g) * o + g * (go * o + bo);
}

__global__ void cross_combine_k(float* __restrict__ hm, float* __restrict__ om,
                                const float* __restrict__ gate,
                                const float* __restrict__ t1,
                                const float* __restrict__ t2) {
  const int total = BTOK * DMD;
  int idx = blockIdx.x * blockDim.x + threadIdx.x;
  if (idx >= total) return;
  const float g = gate[idx];
  const float h = hm[idx] + g * t1[idx];
  const float o = om[idx] + g * t2[idx];
  hm[idx] = h;
  om[idx] = o;
}

// ---------------------------------------------------------------------------
// Host orchestration
// ---------------------------------------------------------------------------
static inline void gemm(int act, const float* X, const float* W, const float* b,
                        float* Y, int M, int N, int K, hipStream_t s) {
  dim3 g(N / 64, M / 16), blk(32);   // M%16==0, N%64==0, K%4==0 guaranteed
  switch (act) {
    case ACT_RELU:    gemm_xwT_wmma<ACT_RELU><<<g, blk, 0, s>>>(X, W, b, Y, N, K); break;
    case ACT_GELU:    gemm_xwT_wmma<ACT_GELU><<<g, blk, 0, s>>>(X, W, b, Y, N, K); break;
    case ACT_SIGMOID: gemm_xwT_wmma<ACT_SIGMOID><<<g, blk, 0, s>>>(X, W, b, Y, N, K); break;
    default:          gemm_xwT_wmma<ACT_NONE><<<g, blk, 0, s>>>(X, W, b, Y, N, K); break;
  }
}

struct TLP {  // transformer layer params, leaf order: W1,W2,Wo,Wqkv,b1,b1f,b2,b2f,bo,bqkv,g1,g2
  const float *W1, *W2, *Wo, *Wqkv, *b1, *b1f, *b2, *b2f, *bo, *bqkv, *g1, *g2;
};

static void run_tlayer(const TLP& p, float* x, int Dm, int nhead, int ff,
                       float* qkvbuf, float* attbuf, float* ffbuf, hipStream_t s) {
  gemm(ACT_NONE, x, p.Wqkv, p.bqkv, qkvbuf, BTOK, 3 * Dm, Dm, s);
  const int tot = 64 * nhead * 4;
  attn_small_k<<<(tot + 127) / 128, 128, 0, s>>>(qkvbuf, attbuf, Dm, nhead);
  gemm(ACT_NONE, attbuf, p.Wo, p.bo, qkvbuf, BTOK, Dm, Dm, s);  // o2 -> reuse qkv
  ln_res_k<<<BTOK, 256, 0, s>>>(x, qkvbuf, p.g1, p.b1, x, Dm);
  gemm(ACT_GELU, x, p.W1, p.b1f, ffbuf, BTOK, ff, Dm, s);
  gemm(ACT_NONE, ffbuf, p.W2, p.b2f, qkvbuf, BTOK, Dm, ff, s);
  ln_res_k<<<BTOK, 256, 0, s>>>(x, qkvbuf, p.g2, p.b2, x, Dm);
}

extern "C" void kernel_launch(void* const* d_in, const int* in_sizes, int n_in,
                              void* d_out, int out_size, void* d_ws, size_t ws_size,
                              hipStream_t stream) {
  (void)in_sizes; (void)n_in; (void)out_size; (void)ws_size;
  // Inputs: dict order for positional args, then params pytree leaves in JAX
  // order (dict keys sorted alphabetically, recursively).
  const float* human_feat = (const float*)d_in[0];
  const float* object_feat = (const float*)d_in[1];
  const float* hxyz = (const float*)d_in[2];
  const float* oxyz = (const float*)d_in[3];
  const float* ov2d = (const float*)d_in[4];
  const float* Km   = (const float*)d_in[5];
  const float* bbox = (const float*)d_in[6];
  const float* cgate_b = (const float*)d_in[7];
  const float* cgate_w = (const float*)d_in[8];
  const float* chfo_b = (const float*)d_in[9];
  const float* chfo_w = (const float*)d_in[10];
  const float* cofh_b = (const float*)d_in[11];
  const float* cofh_w = (const float*)d_in[12];
  const float* conv1_b = (const float*)d_in[13];
  const float* conv1_w = (const float*)d_in[14];
  const float* conv2_b = (const float*)d_in[15];
  const float* conv2_w = (const float*)d_in[16];
  const float* conv3_b = (const float*)d_in[17];
  const float* conv3_w = (const float*)d_in[18];
  const float* conv4_b = (const float*)d_in[19];
  const float* conv4_w = (const float*)d_in[20];
  const float* filmh_b = (const float*)d_in[21];
  const float* filmh_w = (const float*)d_in[22];
  const float* filmo_b = (const float*)d_in[23];
  const float* filmo_w = (const float*)d_in[24];
  const float* imlp_b1 = (const float*)d_in[25];
  const float* imlp_b2 = (const float*)d_in[26];
  const float* imlp_w1 = (const float*)d_in[27];
  const float* imlp_w2 = (const float*)d_in[28];
  const float* iproj_b = (const float*)d_in[29];
  const float* iproj_w = (const float*)d_in[30];
  const float* nh_b = (const float*)d_in[31];
  const float* nh_g = (const float*)d_in[32];
  const float* no_b = (const float*)d_in[33];
  const float* no_g = (const float*)d_in[34];
  const float* proj_b = (const float*)d_in[35];
  const float* proj_w = (const float*)d_in[36];
  auto layer = [&](int base) {
    TLP t;
    t.W1 = (const float*)d_in[base + 0];  t.W2 = (const float*)d_in[base + 1];
    t.Wo = (const float*)d_in[base + 2];  t.Wqkv = (const float*)d_in[base + 3];
    t.b1 = (const float*)d_in[base + 4];  t.b1f = (const float*)d_in[base + 5];
    t.b2 = (const float*)d_in[base + 6];  t.b2f = (const float*)d_in[base + 7];
    t.bo = (const float*)d_in[base + 8];  t.bqkv = (const float*)d_in[base + 9];
    t.g1 = (const float*)d_in[base + 10]; t.g2 = (const float*)d_in[base + 11];
    return t;
  };
  const TLP th = layer(37), ti = layer(49), to = layer(61);

  // workspace arena (~70 MB of floats)
  float* p = (float*)d_ws;
  size_t off = 0;
  auto wsa = [&](size_t n) { float* r = p + off; off += n; return r; };
  float* convA = wsa(9437184);   // conv ping (256,16,48,48) / conv3 out
  float* convB = wsa(4718592);   // conv pong (256,32,24,24) / conv4 out
  float* poolf = wsa(BTOK * 128);
  float* emb   = wsa(BTOK * 128);
  float* s_h   = wsa(BTOK * NHP);
  float* dminh = wsa(BTOK * NHP);
  int*   idxo  = (int*)wsa(BTOK * NHP);
  float* feats = wsa(BTOK * 10);
  float* feats12 = wsa(BTOK * 12);
  float* w1pad = wsa(64 * 12);
  float* hid64 = wsa(BTOK * 64);
  float* igeo  = wsa(BTOK * 128);
  float* catb  = wsa(BTOK * 256);
  float* inter = wsa(BTOK * INTERD);
  float* qkvb  = wsa(BTOK * 3 * DMD);
  float* attb  = wsa(BTOK * DMD);
  float* ffb   = wsa(BTOK * 2048);
  float* fh    = wsa(BTOK * 2 * DMD);
  float* fo    = wsa(BTOK * 2 * DMD);
  float* gate  = wsa(BTOK * DMD);
  float* hmod  = wsa(BTOK * DMD);
  float* omod  = wsa(BTOK * DMD);
  float* tmp1  = wsa(BTOK * DMD);
  float* tmp2  = wsa(BTOK * DMD);
  float* ctr   = wsa(BTOK * 3);
  float* amag  = wsa(BTOK);
  float* gbad  = wsa(BTOK);

  // --- OverlapEncoder ---
  conv1_relu_k<<<(BTOK * 16 * 48 * 48 + 255) / 256, 256, 0, stream>>>(ov2d, conv1_w, conv1_b, convA);
  conv3x3_s2_relu_k<<<(BTOK * 32 * 24 * 24 + 255) / 256, 256, 0, stream>>>(convA, conv2_w, conv2_b, convB, 16, 32, 48, 24);
  conv3x3_s2_relu_k<<<(BTOK * 64 * 12 * 12 + 255) / 256, 256, 0, stream>>>(convB, conv3_w, conv3_b, convA, 32, 64, 24, 12);
  conv3x3_s2_relu_k<<<(BTOK * 128 * 6 * 6 + 255) / 256, 256, 0, stream>>>(convA, conv4_w, conv4_b, convB, 64, 128, 12, 6);
  avgpool36_k<<<(BTOK * 128 + 255) / 256, 256, 0, stream>>>(convB, poolf);
  gemm(ACT_NONE, poolf, proj_w, proj_b, emb, BTOK, 128, 128, stream);

  // --- geometry features ---
  proj_sample_k<<<(BTOK * NHP + 255) / 256, 256, 0, stream>>>(hxyz, ov2d, Km, bbox, s_h, NHP);
  rowmin_k<<<BTOK, 256, 0, stream>>>(hxyz, oxyz, dminh, idxo);
  colmin_mean_k<<<BTOK, 256, 0, stream>>>(hxyz, oxyz, feats);
  feats_k<<<BTOK, 256, 0, stream>>>(dminh, idxo, s_h, hxyz, oxyz, feats);
  pad_rows_k<<<(BTOK * 12 + 255) / 256, 256, 0, stream>>>(feats, feats12, BTOK, 10, 12);
  pad_rows_k<<<(64 * 12 + 255) / 256, 256, 0, stream>>>(imlp_w1, w1pad, 64, 10, 12);
  gemm(ACT_RELU, feats12, w1pad, imlp_b1, hid64, BTOK, 64, 12, stream);
  gemm(ACT_NONE, hid64, imlp_w2, imlp_b2, igeo, BTOK, 128, 64, stream);

  // --- interaction fusion ---
  concat_k<<<(BTOK * 256 + 255) / 256, 256, 0, stream>>>(emb, igeo, catb);
  gemm(ACT_NONE, catb, iproj_w, iproj_b, inter, BTOK, INTERD, 256, stream);
  run_tlayer(ti, inter, INTERD, 4, 1024, qkvb, attb, ffb, stream);

  gemm(ACT_NONE, inter, filmh_w, filmh_b, fh, BTOK, 2 * DMD, INTERD, stream);
  gemm(ACT_NONE, inter, filmo_w, filmo_b, fo, BTOK, 2 * DMD, INTERD, stream);
  gemm(ACT_SIGMOID, inter, cgate_w, cgate_b, gate, BTOK, DMD, INTERD, stream);

  centers_k<<<(BTOK * 3 + 255) / 256, 256, 0, stream>>>(oxyz, ctr);
  amag_k<<<BTOK, 256, 0, stream>>>(oxyz, ctr, amag);
  gbad_k<<<4, 64, 0, stream>>>(amag, gbad);

  film_mod_k<<<(BTOK * DMD + 255) / 256, 256, 0, stream>>>(human_feat, object_feat, fh, fo, gbad, hmod, omod);
  gemm(ACT_NONE, omod, chfo_w, chfo_b, tmp1, BTOK, DMD, DMD, stream);
  gemm(ACT_NONE, hmod, cofh_w, cofh_b, tmp2, BTOK, DMD, DMD, stream);
  cross_combine_k<<<(BTOK * DMD + 255) / 256, 256, 0, stream>>>(hmod, omod, gate, tmp1, tmp2);

  run_tlayer(th, hmod, DMD, 8, 2048, qkvb, attb, ffb, stream);
  run_tlayer(to, omod, DMD, 8, 2048, qkvb, attb, ffb, stream);

  float* outp = (float*)d_out;
  ln_res_k<<<BTOK, 256, 0, stream>>>(hmod, nullptr, nh_g, nh_b, outp, DMD);
  ln_res_k<<<BTOK, 256, 0, stream>>>(omod, nullptr, no_g, no_b, outp + (size_t)BTOK * DMD, DMD);
}